// SOM_49426483642930
// MI455X (gfx1250) — compile-verified
//
#include <hip/hip_runtime.h>
#include <math.h>

// ---------------------------------------------------------------------------
// SOM scan, restructured:
//   W_b = W0 + off_b[c]  (per-cell scalar cumsum)  =>
//   Phase 1: persistent kernel, 81 blocks (one per map cell), W0 cell cached
//            in registers, 128 sequential steps synced by a software grid
//            barrier; emits cumulative gain table G[128][81].
//   Phase 2: out[b,c,:,:] = W0[c,:,:] + G[b][c] via V_WMMA_F32_16X16X4_F32
//            (A col0 = g, B row0 = 1 -> D = g + C). Store-BW bound.
// ---------------------------------------------------------------------------

#define NCELL 81
#define CELL_ELEMS 16384   // 128*128
#define NSTEP 128

struct SomCtl {
    unsigned arrive;
    unsigned step;
    unsigned pad0;
    unsigned pad1;
    float    off[NCELL];
    float    rings[NCELL];
    float    minv[NCELL];
    unsigned minp[NCELL];
    float    G[NSTEP * NCELL];   // cumulative offsets after each step
};

// exp(-0.01) in f64, truncated to f32 as the reference does
#define LR_F     0.99004983374916805f
#define RADIUS_F 4.9502491687458403f

__global__ void som_init(SomCtl* ws) {
    int t = threadIdx.x;
    if (t == 0) { ws->arrive = 0u; ws->step = 0u; }
    if (t < NCELL) ws->off[t] = 0.0f;
}

__global__ __launch_bounds__(256) void som_phase1(const float* __restrict__ x,
                                                  const float* __restrict__ W0,
                                                  SomCtl* ws) {
    const int c   = blockIdx.x;       // cell 0..80
    const int tid = threadIdx.x;      // 0..255

    // Cache my 64KB cell of W0 in registers: 16 x float4 per thread.
    float4 wreg[16];
    const float4* wv = (const float4*)(W0 + (size_t)c * CELL_ELEMS);
#pragma unroll
    for (int k = 0; k < 16; ++k) wreg[k] = wv[k * 256 + tid];

    __shared__ float    s_sum[256];
    __shared__ float    s_val[256];
    __shared__ unsigned s_idx[256];
    __shared__ float    s_off;

    float off = 0.0f;
    const float TWO_R2 = 2.0f * RADIUS_F * RADIUS_F;

    for (int b = 0; b < NSTEP; ++b) {
        const float4* xv = (const float4*)(x + (size_t)b * CELL_ELEMS);
        float sum = 0.0f;
        float mval = __builtin_huge_valf();
        unsigned mp = 0u;
#pragma unroll
        for (int k = 0; k < 16; ++k) {
            float4 xd = xv[k * 256 + tid];
            unsigned pbase = (unsigned)((k * 256 + tid) * 4);
            float d;
            d = fabsf(wreg[k].x + off - xd.x); sum += d;
            if (d < mval) { mval = d; mp = pbase + 0u; }
            d = fabsf(wreg[k].y + off - xd.y); sum += d;
            if (d < mval) { mval = d; mp = pbase + 1u; }
            d = fabsf(wreg[k].z + off - xd.z); sum += d;
            if (d < mval) { mval = d; mp = pbase + 2u; }
            d = fabsf(wreg[k].w + off - xd.w); sum += d;
            if (d < mval) { mval = d; mp = pbase + 3u; }
        }

        // block reduction: sum and lexicographic (value, index) min
        s_sum[tid] = sum; s_val[tid] = mval; s_idx[tid] = mp;
        __syncthreads();
        for (int s = 128; s >= 1; s >>= 1) {
            if (tid < s) {
                s_sum[tid] += s_sum[tid + s];
                float    v2 = s_val[tid + s];
                unsigned p2 = s_idx[tid + s];
                if (v2 < s_val[tid] || (v2 == s_val[tid] && p2 < s_idx[tid])) {
                    s_val[tid] = v2; s_idx[tid] = p2;
                }
            }
            __syncthreads();
        }

        if (tid == 0) {
            ws->rings[c] = s_sum[0];
            ws->minv[c]  = s_val[0];
            ws->minp[c]  = s_idx[0];
            __threadfence();
            unsigned t = atomicAdd(&ws->arrive, 1u);
            if (t == (unsigned)(NCELL - 1)) {
                // last block to arrive combines & publishes the step
                __threadfence();
                float    bv = ws->minv[0];
                unsigned bc = 0u, bp = ws->minp[0];
                for (unsigned cc = 1; cc < NCELL; ++cc) {
                    float v = ws->minv[cc];
                    if (v < bv) { bv = v; bc = cc; bp = ws->minp[cc]; }
                }
                int flat = (int)(bc * (unsigned)CELL_ELEMS + bp);
                int wx = flat / 9;
                int wy = flat - wx * 9;
                for (int cc = 0; cc < NCELL; ++cc) {
                    int i = cc / 9, j = cc - (cc / 9) * 9;
                    float fx = (float)(i - wx);
                    float fy = (float)(j - wy);
                    float d2 = fx * fx + fy * fy;
                    float dist = sqrtf(d2);
                    float gauss = expf(-(dist * dist) / TWO_R2);
                    float g = (LR_F * gauss) * ws->rings[cc];
                    g = (g < RADIUS_F) ? g : 0.0f;
                    float no = ws->off[cc] + g;
                    ws->off[cc] = no;
                    ws->G[b * NCELL + cc] = no;
                }
                __threadfence();
                ws->arrive = 0u;
                __threadfence();
                __hip_atomic_store(&ws->step, (unsigned)(b + 1),
                                   __ATOMIC_RELEASE, __HIP_MEMORY_SCOPE_AGENT);
            } else {
                while (__hip_atomic_load(&ws->step, __ATOMIC_ACQUIRE,
                                         __HIP_MEMORY_SCOPE_AGENT) < (unsigned)(b + 1)) {
                    __builtin_amdgcn_s_sleep(2);
                }
            }
            s_off = ws->off[c];   // == G[b][c]
        }
        __syncthreads();
        off = s_off;
    }
}

typedef float v2f __attribute__((ext_vector_type(2)));
typedef float v8f __attribute__((ext_vector_type(8)));

__global__ __launch_bounds__(256) void som_phase2(const float* __restrict__ W0,
                                                  const SomCtl* __restrict__ ws,
                                                  float* __restrict__ out) {
    const int c    = blockIdx.x;      // cell  0..80
    const int b    = blockIdx.y;      // batch 0..127
    const int tid  = threadIdx.x;
    const int wave = tid >> 5;        // 0..7 -> 16-row strip
    const int lane = tid & 31;
    const int half = lane >> 4;       // 0: rows r, 1: rows r+8
    const int n    = lane & 15;       // column within 16-wide tile

    const float g = ws->G[b * NCELL + c];
    const float* wbase = W0 + (size_t)c * CELL_ELEMS;
    float* obase = out + ((size_t)b * NCELL + (size_t)c) * CELL_ELEMS;

    // A(16x4): column 0 = g, rest 0.   B(4x16): row 0 = ones, rest 0.
    // => A*B = g * ones(16,16);  D = g + C  (exact fp32 add, RNE).
    v2f a;  a.x  = half ? 0.0f : g;    a.y  = 0.0f;
    v2f bb; bb.x = half ? 0.0f : 1.0f; bb.y = 0.0f;

    const int row0 = wave * 16 + half * 8;
#pragma unroll
    for (int tn = 0; tn < 8; ++tn) {
        const int col = tn * 16 + n;
        v8f cacc;
#pragma unroll
        for (int r = 0; r < 8; ++r) cacc[r] = wbase[(row0 + r) * 128 + col];
        v8f d = __builtin_amdgcn_wmma_f32_16x16x4_f32(
            /*neg_a=*/false, a, /*neg_b=*/false, bb,
            /*c_mod=*/(short)0, cacc, /*reuse_a=*/false, /*reuse_b=*/false);
#pragma unroll
        for (int r = 0; r < 8; ++r) obase[(row0 + r) * 128 + col] = d[r];
    }
}

extern "C" void kernel_launch(void* const* d_in, const int* in_sizes, int n_in,
                              void* d_out, int out_size, void* d_ws, size_t ws_size,
                              hipStream_t stream) {
    const float* x  = (const float*)d_in[0];   // (128,128,128) f32
    const float* W0 = (const float*)d_in[1];   // (9,9,128,128) f32
    float* out = (float*)d_out;                // (128,9,9,128,128) f32
    SomCtl* ws = (SomCtl*)d_ws;                // needs ~43 KB scratch

    som_init<<<1, 128, 0, stream>>>(ws);
    som_phase1<<<NCELL, 256, 0, stream>>>(x, W0, ws);
    som_phase2<<<dim3(NCELL, NSTEP), 256, 0, stream>>>(W0, ws, out);
}